// MultiBoxLoss_23364622090954
// MI455X (gfx1250) — compile-verified
//
#include <hip/hip_runtime.h>
#include <hip/hip_bf16.h>

#define BB 64
#define PP 16384
#define CC 81
#define NOBJ 32

typedef __attribute__((ext_vector_type(2))) float v2f;
typedef __attribute__((ext_vector_type(8))) float v8f;

__device__ __forceinline__ float waveMax(float v) {
    #pragma unroll
    for (int m = 16; m > 0; m >>= 1) v = fmaxf(v, __shfl_xor(v, m, 32));
    return v;
}
__device__ __forceinline__ float waveSum(float v) {
    #pragma unroll
    for (int m = 16; m > 0; m >>= 1) v += __shfl_xor(v, m, 32);
    return v;
}

// ---------------- init: zero argmax keys + n_pos ----------------
__global__ void mb_init_kernel(unsigned long long* gkeys, int* n_pos) {
    int i = blockIdx.x * blockDim.x + threadIdx.x;
    if (i < BB * NOBJ) gkeys[i] = 0ULL;
    if (i < BB) n_pos[i] = 0;
}

// ---------------- kernel 1: IoU matching ----------------
// grid = B * (P/256), block = 256; one thread per prior
__global__ void mb_match_kernel(const float* __restrict__ boxes,
                                const float* __restrict__ priors,
                                float* __restrict__ ov_out,
                                int* __restrict__ obj_out,
                                unsigned long long* __restrict__ gkeys) {
    __shared__ float sb[NOBJ][4];
    __shared__ float sarea[NOBJ];
    __shared__ unsigned long long lkeys[NOBJ];

    int b = blockIdx.x >> 6;          // 16384/256 = 64 blocks per batch
    int pblk = blockIdx.x & 63;
    int p = pblk * 256 + threadIdx.x;

    if (threadIdx.x < NOBJ) {
        int o = threadIdx.x;
        const float* bx = boxes + ((size_t)b * NOBJ + o) * 4;
        sb[o][0] = bx[0]; sb[o][1] = bx[1]; sb[o][2] = bx[2]; sb[o][3] = bx[3];
        sarea[o] = (bx[2] - bx[0]) * (bx[3] - bx[1]);
        lkeys[o] = 0ULL;
    }
    __syncthreads();

    const float* pr = priors + (size_t)p * 4;
    float cx = pr[0], cy = pr[1], w = pr[2], h = pr[3];
    float px0 = cx - w * 0.5f, py0 = cy - h * 0.5f;
    float px1 = cx + w * 0.5f, py1 = cy + h * 0.5f;
    float parea = (px1 - px0) * (py1 - py0);

    float best = -1.0f; int besto = 0;
    #pragma unroll 4
    for (int o = 0; o < NOBJ; ++o) {
        float lx = fmaxf(sb[o][0], px0), ly = fmaxf(sb[o][1], py0);
        float hx = fminf(sb[o][2], px1), hy = fminf(sb[o][3], py1);
        float iw = fmaxf(hx - lx, 0.0f), ih = fmaxf(hy - ly, 0.0f);
        float inter = iw * ih;
        float iou = inter / (sarea[o] + parea - inter);
        if (iou > best) { best = iou; besto = o; }   // first-wins tie-break (argmax)
        // key: value-major, then smaller prior index wins ties (~p)
        unsigned long long key =
            ((unsigned long long)__float_as_uint(iou) << 32) | (unsigned int)(~p);
        atomicMax(&lkeys[o], key);
    }
    size_t idx = (size_t)b * PP + p;
    ov_out[idx] = best;
    obj_out[idx] = besto;
    __syncthreads();
    if (threadIdx.x < NOBJ)
        atomicMax(&gkeys[b * NOBJ + threadIdx.x], lkeys[threadIdx.x]);
}

// ---------------- kernel 2: force each object to own its best prior ----------------
// sequential over objects per batch -> "last wins" scatter like the reference
__global__ void mb_override_kernel(const unsigned long long* __restrict__ gkeys,
                                   float* __restrict__ ov,
                                   int* __restrict__ obj) {
    int b = blockIdx.x * blockDim.x + threadIdx.x;
    if (b >= BB) return;
    for (int o = 0; o < NOBJ; ++o) {
        unsigned long long key = gkeys[b * NOBJ + o];
        unsigned int p = ~(unsigned int)(key & 0xFFFFFFFFu);
        size_t idx = (size_t)b * PP + p;
        obj[idx] = o;
        ov[idx] = 1.0f;
    }
}

// ---------------- kernel 3: per-anchor CE + positive L1 loc loss ----------------
// grid = B*P/256 = 4096, block = 256 (8 waves); each wave handles 32 anchors,
// lane = class c, c+32, c+64 -> fully coalesced stream of the 340MB score tensor.
__global__ void mb_ce_loc_kernel(const float* __restrict__ locs,
                                 const float* __restrict__ scores,
                                 const float* __restrict__ boxes,
                                 const int* __restrict__ labels,
                                 const float* __restrict__ priors,
                                 const float* __restrict__ ov,
                                 const int* __restrict__ obj,
                                 float* __restrict__ ce_neg,
                                 int* __restrict__ n_pos,
                                 float* __restrict__ cepos_part,
                                 float* __restrict__ loc_part) {
    int lane = threadIdx.x & 31;
    int wid = threadIdx.x >> 5;
    int b = blockIdx.x >> 6;                    // 64 blocks per batch
    int blockAnchor = blockIdx.x * 256;

    float ce_acc = 0.0f, loc_acc = 0.0f;
    int np_acc = 0;

    for (int it = 0; it < 32; ++it) {
        int a = blockAnchor + wid * 32 + it;    // global anchor = b*P + p
        int p = a & (PP - 1);
        const float* srow = scores + (size_t)a * CC;

        float s0 = srow[lane];
        float s1 = srow[32 + lane];
        float s2 = (lane < 17) ? srow[64 + lane] : -3.0e38f;

        float m = fmaxf(s0, fmaxf(s1, s2));
        m = waveMax(m);
        float e = __expf(s0 - m) + __expf(s1 - m) +
                  ((lane < 17) ? __expf(s2 - m) : 0.0f);
        float sum = waveSum(e);

        int ob = obj[a];
        float ovv = ov[a];
        int lab = (ovv < 0.5f) ? 0 : labels[b * NOBJ + ob];

        // fetch score of the true class from registers via cross-lane read
        float src  = (lab < 32) ? s0 : ((lab < 64) ? s1 : s2);
        int   srcl = (lab < 32) ? lab : ((lab < 64) ? (lab - 32) : (lab - 64));
        float slab = __shfl(src, srcl, 32);

        float ce = __logf(sum) + m - slab;
        bool pos = (lab != 0);

        if (lane == 0) {
            ce_neg[a] = pos ? 0.0f : ce;
            if (pos) {
                ce_acc += ce;
                np_acc += 1;
                const float* bx = boxes + ((size_t)b * NOBJ + ob) * 4;
                float bcx = (bx[0] + bx[2]) * 0.5f;
                float bcy = (bx[1] + bx[3]) * 0.5f;
                float bw = bx[2] - bx[0];
                float bh = bx[3] - bx[1];
                const float* pc = priors + (size_t)p * 4;
                float gx = (bcx - pc[0]) / (pc[2] / 10.0f);
                float gy = (bcy - pc[1]) / (pc[3] / 10.0f);
                float gw = __logf(bw / pc[2]) * 5.0f;
                float gh = __logf(bh / pc[3]) * 5.0f;
                const float* pl = locs + (size_t)a * 4;
                loc_acc += fabsf(pl[0] - gx) + fabsf(pl[1] - gy) +
                           fabsf(pl[2] - gw) + fabsf(pl[3] - gh);
            }
        }
    }

    __shared__ float sce[8], sloc[8];
    __shared__ int snp[8];
    if (lane == 0) { sce[wid] = ce_acc; sloc[wid] = loc_acc; snp[wid] = np_acc; }
    __syncthreads();
    if (threadIdx.x == 0) {
        float tc = 0.0f, tl = 0.0f; int tn = 0;
        #pragma unroll
        for (int i = 0; i < 8; ++i) { tc += sce[i]; tl += sloc[i]; tn += snp[i]; }
        cepos_part[blockIdx.x] = tc;
        loc_part[blockIdx.x] = tl;
        atomicAdd(&n_pos[b], tn);
    }
}

// ---------------- kernel 4: hard-negative mining via in-LDS bitonic sort ----------------
// one block per batch row; 16384 floats = 64KB in the WGP's 320KB LDS
__global__ void __launch_bounds__(1024)
mb_hardneg_kernel(const float* __restrict__ ce_neg,
                  const int* __restrict__ n_pos,
                  float* __restrict__ neg_part) {
    __shared__ float s[PP];
    int b = blockIdx.x;
    const float* row = ce_neg + (size_t)b * PP;
    for (int i = threadIdx.x; i < PP; i += blockDim.x) s[i] = row[i];
    __syncthreads();

    for (int k = 2; k <= PP; k <<= 1) {
        for (int j = k >> 1; j > 0; j >>= 1) {
            for (int i = threadIdx.x; i < PP; i += blockDim.x) {
                int ixj = i ^ j;
                if (ixj > i) {
                    float x = s[i], y = s[ixj];
                    bool descBlk = ((i & k) == 0);
                    if (descBlk ? (x < y) : (x > y)) { s[i] = y; s[ixj] = x; }
                }
            }
            __syncthreads();
        }
    }

    int topk = 3 * n_pos[b];
    if (topk > PP) topk = PP;
    float acc = 0.0f;
    for (int i = threadIdx.x; i < topk; i += blockDim.x) acc += s[i];
    __syncthreads();
    s[threadIdx.x] = acc;
    __syncthreads();
    for (int off = blockDim.x >> 1; off > 0; off >>= 1) {
        if (threadIdx.x < (unsigned)off) s[threadIdx.x] += s[threadIdx.x + off];
        __syncthreads();
    }
    if (threadIdx.x == 0) neg_part[b] = s[0];
}

// ---------------- kernel 5: WMMA-based deterministic final reduction ----------------
// D = A(16x4) x ones(4x16) + C  -> each WMMA folds 64 partials into row-sums.
__device__ __forceinline__ v8f wmma_fold(const float* __restrict__ x, int n, v8f c) {
    v2f ones; ones.x = 1.0f; ones.y = 1.0f;
    int lane = threadIdx.x & 31;
    int r = (lane < 16) ? lane : (lane - 16);
    int koff = (lane < 16) ? 0 : 2;
    for (int base = 0; base < n; base += 64) {
        v2f a;
        a.x = x[base + 4 * r + koff];
        a.y = x[base + 4 * r + koff + 1];
        c = __builtin_amdgcn_wmma_f32_16x16x4_f32(false, a, false, ones,
                                                  (short)0, c, false, false);
    }
    return c;
}

__device__ __forceinline__ float wmma_total(v8f c) {
    float t = 0.0f;
    #pragma unroll
    for (int i = 0; i < 8; ++i) t += c[i];
    // all 16 columns carry identical row-sums -> wave total = 16 * grand total
    return waveSum(t) * 0.0625f;
}

__global__ void mb_finalize_kernel(const float* __restrict__ cepos_part,
                                   const float* __restrict__ loc_part,
                                   const float* __restrict__ neg_part,
                                   const int* __restrict__ n_pos,
                                   float* __restrict__ out) {
    v8f c0 = {0.f,0.f,0.f,0.f,0.f,0.f,0.f,0.f};
    v8f c1 = {0.f,0.f,0.f,0.f,0.f,0.f,0.f,0.f};
    v8f c2 = {0.f,0.f,0.f,0.f,0.f,0.f,0.f,0.f};
    c0 = wmma_fold(cepos_part, 4096, c0);
    c1 = wmma_fold(loc_part,   4096, c1);
    c2 = wmma_fold(neg_part,     64, c2);
    float cepos  = wmma_total(c0);
    float locsum = wmma_total(c1);
    float negsum = wmma_total(c2);

    int tn = 0;
    for (int i = 0; i < BB; ++i) tn += n_pos[i];   // uniform scalar loop
    float total_pos = (float)tn;

    float loss = (cepos + negsum) / total_pos + locsum / (total_pos * 4.0f);
    if ((threadIdx.x & 31) == 0) out[0] = loss;
}

// ---------------- host launch ----------------
extern "C" void kernel_launch(void* const* d_in, const int* in_sizes, int n_in,
                              void* d_out, int out_size, void* d_ws, size_t ws_size,
                              hipStream_t stream) {
    const float* locs   = (const float*)d_in[0];   // [B,P,4]
    const float* scores = (const float*)d_in[1];   // [B,P,81]
    const float* boxes  = (const float*)d_in[2];   // [B,32,4] xy
    const int*   labels = (const int*)  d_in[3];   // [B,32]
    const float* priors = (const float*)d_in[4];   // [P,4] cxcy

    unsigned char* ws = (unsigned char*)d_ws;
    const size_t BP = (size_t)BB * PP;
    float*              ov      = (float*)(ws);                       // 4 MB
    int*                obj     = (int*)  (ws + BP * 4);              // 4 MB
    float*              ce_neg  = (float*)(ws + BP * 8);              // 4 MB
    unsigned long long* gkeys   = (unsigned long long*)(ws + BP * 12);        // 16 KB
    int*                n_pos   = (int*)  (ws + BP * 12 + 16384);             // 256 B
    float*              cepos_p = (float*)(ws + BP * 12 + 16384 + 256);       // 16 KB
    float*              loc_p   = (float*)(ws + BP * 12 + 32768 + 256);       // 16 KB
    float*              neg_p   = (float*)(ws + BP * 12 + 49152 + 256);       // 256 B

    mb_init_kernel<<<(BB * NOBJ + 255) / 256, 256, 0, stream>>>(gkeys, n_pos);

    mb_match_kernel<<<BB * (PP / 256), 256, 0, stream>>>(boxes, priors, ov, obj, gkeys);

    mb_override_kernel<<<1, 64, 0, stream>>>(gkeys, ov, obj);

    mb_ce_loc_kernel<<<BB * (PP / 256), 256, 0, stream>>>(
        locs, scores, boxes, labels, priors, ov, obj, ce_neg, n_pos, cepos_p, loc_p);

    mb_hardneg_kernel<<<BB, 1024, 0, stream>>>(ce_neg, n_pos, neg_p);

    mb_finalize_kernel<<<1, 32, 0, stream>>>(cepos_p, loc_p, neg_p, n_pos, (float*)d_out);
}